// PromptGetter_66975720014396
// MI455X (gfx1250) — compile-verified
//
#include <hip/hip_runtime.h>
#include <stdint.h>

#define EMB_C 256
#define SP 4096              // 64*64 spatial
#define NLAB 16
#define OH 1080
#define OW 1920
#define CELL 120             // grid cell = 120x120 pixels (exact, see analysis)
#define GXN 16
#define NGRID 256
#define DEFAULT_THR 0.65f

typedef float v2f __attribute__((ext_vector_type(2)));
typedef float v8f __attribute__((ext_vector_type(8)));

// ---- order-preserving float <-> uint mapping (monotonic) ----
__device__ __forceinline__ unsigned int ford(float f) {
  unsigned int u = __float_as_uint(f);
  return (u & 0x80000000u) ? ~u : (u | 0x80000000u);
}
__device__ __forceinline__ float ford_inv(unsigned int o) {
  unsigned int u = (o & 0x80000000u) ? (o & 0x7FFFFFFFu) : ~o;
  return __uint_as_float(u);
}

// ---------------------------------------------------------------------------
// Kernel 1: per-position inverse channel norm  (tf = emb / ||emb||_c)
// ---------------------------------------------------------------------------
__global__ void pg_norm(const float* __restrict__ emb, float* __restrict__ invn) {
  int p = blockIdx.x * blockDim.x + threadIdx.x;   // 0..4095
  float s = 0.0f;
  #pragma unroll 8
  for (int c = 0; c < EMB_C; ++c) {
    float v = emb[c * SP + p];
    s += v * v;
  }
  invn[p] = 1.0f / sqrtf(s);
}

// ---------------------------------------------------------------------------
// Kernel 2: sim[16][4096] = ref[16][256] @ (emb * invn)  via V_WMMA_F32_16X16X4_F32
// One wave handles 64 spatial columns (4 N-tiles of 16), K looped in steps of 4.
// A (16x4 f32): lanes 0-15 hold M=lane {K=k0,k0+1}; lanes 16-31 hold {K=k0+2,k0+3}.
// B (4x16 f32): lanes 0-15 N=lane {K rows k0,k0+1}; lanes 16-31 {k0+2,k0+3}.
// ---------------------------------------------------------------------------
__global__ __launch_bounds__(256) void pg_gemm(const float* __restrict__ emb,
                                               const float* __restrict__ ref,
                                               const float* __restrict__ invn,
                                               float* __restrict__ sim) {
  const int lane = threadIdx.x & 31;
  const int wv   = (blockIdx.x * blockDim.x + threadIdx.x) >> 5;  // global wave id
  const int p0   = wv * 64;                                       // 64 columns per wave
  const int col  = lane & 15;
  const int koff = (lane < 16) ? 0 : 2;

  v8f acc0 = {}, acc1 = {}, acc2 = {}, acc3 = {};
  const float in0 = invn[p0 + col];
  const float in1 = invn[p0 + 16 + col];
  const float in2 = invn[p0 + 32 + col];
  const float in3 = invn[p0 + 48 + col];

  const float* __restrict__ arow  = ref + col * EMB_C + koff;   // ref is (16,1,256)
  const float* __restrict__ bbase = emb + p0 + col;

  #pragma unroll 4
  for (int k0 = 0; k0 < EMB_C; k0 += 4) {
    v2f a;
    a.x = arow[k0];
    a.y = arow[k0 + 1];
    const float* bp = bbase + (size_t)(k0 + koff) * SP;
    v2f b0, b1, b2, b3;
    b0.x = bp[0]       * in0;  b0.y = bp[SP]      * in0;
    b1.x = bp[16]      * in1;  b1.y = bp[SP + 16] * in1;
    b2.x = bp[32]      * in2;  b2.y = bp[SP + 32] * in2;
    b3.x = bp[48]      * in3;  b3.y = bp[SP + 48] * in3;
    acc0 = __builtin_amdgcn_wmma_f32_16x16x4_f32(false, a, false, b0, (short)0, acc0, false, false);
    acc1 = __builtin_amdgcn_wmma_f32_16x16x4_f32(false, a, false, b1, (short)0, acc1, false, false);
    acc2 = __builtin_amdgcn_wmma_f32_16x16x4_f32(false, a, false, b2, (short)0, acc2, false, false);
    acc3 = __builtin_amdgcn_wmma_f32_16x16x4_f32(false, a, false, b3, (short)0, acc3, false, false);
  }

  // D layout: VGPR r, lanes 0-15 -> M=r; lanes 16-31 -> M=r+8
  const int mb = (lane < 16) ? 0 : 8;
  #pragma unroll
  for (int r = 0; r < 8; ++r) {
    int m = mb + r;
    sim[m * SP + p0 + col]      = acc0[r];
    sim[m * SP + p0 + 16 + col] = acc1[r];
    sim[m * SP + p0 + 32 + col] = acc2[r];
    sim[m * SP + p0 + 48 + col] = acc3[r];
  }
}

// ---------------------------------------------------------------------------
// Kernel 3: init reduction slots
// ---------------------------------------------------------------------------
__global__ void pg_init(unsigned long long* gkeys, unsigned long long* bgk) {
  int i = blockIdx.x * blockDim.x + threadIdx.x;
  if (i < NLAB * NGRID) gkeys[i] = 0ULL;
  if (i < NLAB)         bgk[i]   = 0xFFFFFFFFFFFFFFFFULL;
}

// ---------------------------------------------------------------------------
// Fused double-bilinear sampling: sim(64x64) -> m1(576x1024 crop of 1024^2) -> m2(1080x1920)
// jax.image.resize 'linear' upsample == half-pixel-center bilinear with edge clamp.
// ---------------------------------------------------------------------------
__device__ __forceinline__ float m1v(const float* __restrict__ ssim, int y1, int x1) {
  float ys = ((float)y1 + 0.5f) * 0.0625f - 0.5f;   // 64/1024 scale
  float xs = ((float)x1 + 0.5f) * 0.0625f - 0.5f;
  ys = fminf(fmaxf(ys, 0.0f), 63.0f);
  xs = fminf(fmaxf(xs, 0.0f), 63.0f);
  int y0 = (int)ys, x0 = (int)xs;
  int yy = min(y0 + 1, 63), xx = min(x0 + 1, 63);
  float fy = ys - (float)y0, fx = xs - (float)x0;
  float v00 = ssim[(y0 << 6) + x0], v01 = ssim[(y0 << 6) + xx];
  float v10 = ssim[(yy << 6) + x0], v11 = ssim[(yy << 6) + xx];
  float a = v00 + fx * (v01 - v00);
  float b = v10 + fx * (v11 - v10);
  return a + fy * (b - a);
}

// ---------------------------------------------------------------------------
// Kernel 4: one workgroup per (grid cell, label): compute mask values on the fly,
// reduce max(value,~idx) over valid pixels (segment_max + min-index tie-break)
// and min(value,idx) over all pixels (bg point) in LDS; sim tile staged in LDS.
// ---------------------------------------------------------------------------
__global__ __launch_bounds__(256) void pg_reduce(const float* __restrict__ sim,
                                                 const float* __restrict__ thrp,
                                                 unsigned long long* __restrict__ gkeys,
                                                 unsigned long long* __restrict__ bgk) {
  __shared__ float ssim[SP];
  __shared__ unsigned long long smax[256];
  __shared__ unsigned long long smin[256];
  const int tid   = threadIdx.x;
  const int cell  = blockIdx.x;     // 0..143
  const int label = blockIdx.y;     // 0..15
  const int gy = cell >> 4, gx = cell & 15;

  for (int i = tid; i < SP; i += 256) ssim[i] = sim[label * SP + i];
  __syncthreads();

  float tin = thrp[0];
  const float thr = (tin == 0.0f) ? DEFAULT_THR : tin;

  unsigned long long bmax = 0ULL;
  unsigned long long bmin = 0xFFFFFFFFFFFFFFFFULL;
  const int Y0 = gy * CELL, X0 = gx * CELL;

  for (int i = tid; i < CELL * CELL; i += 256) {
    int py = i / CELL, px = i - py * CELL;
    int Y = Y0 + py, X = X0 + px;
    // stage-2 resize coords: 576x1024 -> 1080x1920 (both scales = 8/15)
    float ym = ((float)Y + 0.5f) * (576.0f / 1080.0f) - 0.5f;
    float xm = ((float)X + 0.5f) * (1024.0f / 1920.0f) - 0.5f;
    ym = fminf(fmaxf(ym, 0.0f), 575.0f);
    xm = fminf(fmaxf(xm, 0.0f), 1023.0f);
    int y0 = (int)ym, x0 = (int)xm;
    int y1 = min(y0 + 1, 575), x1 = min(x0 + 1, 1023);
    float fy = ym - (float)y0, fx = xm - (float)x0;
    float v00 = m1v(ssim, y0, x0);
    float v01 = m1v(ssim, y0, x1);
    float v10 = m1v(ssim, y1, x0);
    float v11 = m1v(ssim, y1, x1);
    float va  = v00 + fx * (v01 - v00);
    float vb  = v10 + fx * (v11 - v10);
    float val = va + fy * (vb - va);

    unsigned int idx = (unsigned int)(Y * OW + X);
    unsigned long long kmin = ((unsigned long long)ford(val) << 32) | idx;
    if (kmin < bmin) bmin = kmin;
    if (val > thr) {
      unsigned long long kmax = ((unsigned long long)ford(val) << 32) | (unsigned int)(~idx);
      if (kmax > bmax) bmax = kmax;
    }
  }

  smax[tid] = bmax;
  smin[tid] = bmin;
  __syncthreads();
  for (int s = 128; s > 0; s >>= 1) {
    if (tid < s) {
      if (smax[tid + s] > smax[tid]) smax[tid] = smax[tid + s];
      if (smin[tid + s] < smin[tid]) smin[tid] = smin[tid + s];
    }
    __syncthreads();
  }
  if (tid == 0) {
    gkeys[label * NGRID + (gy * GXN + gx)] = smax[0];
    atomicMin(&bgk[label], smin[0]);
  }
}

// ---------------------------------------------------------------------------
// Kernel 5: per label, decode 256 grid slots, stable-sort descending by score
// (rank counting reproduces jnp.argsort(-score) stability), write outputs.
// ---------------------------------------------------------------------------
__global__ __launch_bounds__(256) void pg_final(const unsigned long long* __restrict__ gkeys,
                                                const unsigned long long* __restrict__ bgk,
                                                float* __restrict__ out) {
  __shared__ float sscore[NGRID];
  const int t = threadIdx.x;
  const int label = blockIdx.x;

  unsigned long long key = gkeys[label * NGRID + t];
  float score, bx, by;
  if (key != 0ULL) {
    score = ford_inv((unsigned int)(key >> 32));
    unsigned int idx = ~(unsigned int)(key & 0xFFFFFFFFULL);
    bx = (float)(idx % OW);
    by = (float)(idx / OW);
  } else {                       // empty grid -> (0, 0, -inf)
    score = -__builtin_inff();
    bx = 0.0f;
    by = 0.0f;
  }
  sscore[t] = score;
  __syncthreads();

  int rank = 0;
  for (int j = 0; j < NGRID; ++j) {
    float sj = sscore[j];
    if (sj > score || (sj == score && j < t)) ++rank;
  }
  float* dst = out + ((size_t)label * NGRID + rank) * 3;
  dst[0] = bx;
  dst[1] = by;
  dst[2] = score;

  if (t == 0) {
    unsigned long long bk = bgk[label];
    unsigned int idx = (unsigned int)(bk & 0xFFFFFFFFULL);
    out[NLAB * NGRID * 3 + label * 2 + 0] = (float)(idx % OW);  // bg_col (x)
    out[NLAB * NGRID * 3 + label * 2 + 1] = (float)(idx / OW);  // bg_row (y)
  }
}

// ---------------------------------------------------------------------------
extern "C" void kernel_launch(void* const* d_in, const int* in_sizes, int n_in,
                              void* d_out, int out_size, void* d_ws, size_t ws_size,
                              hipStream_t stream) {
  (void)in_sizes; (void)n_in; (void)out_size; (void)ws_size;
  const float* emb  = (const float*)d_in[0];   // (1,256,64,64) f32
  const float* ref  = (const float*)d_in[1];   // (16,1,256)    f32
  const float* thrp = (const float*)d_in[3];   // (1,1)         f32
  float* out = (float*)d_out;                  // 16*256*3 + 16*2 f32

  char* ws = (char*)d_ws;
  float* invn = (float*)ws;                                   //  4096 f32
  float* sim  = (float*)(ws + 16384);                         // 65536 f32
  unsigned long long* gkeys = (unsigned long long*)(ws + 16384 + 262144);  // 16*256 u64
  unsigned long long* bgk   = gkeys + NLAB * NGRID;                        // 16 u64

  pg_norm<<<SP / 256, 256, 0, stream>>>(emb, invn);
  pg_init<<<(NLAB * NGRID + 255) / 256 + 1, 256, 0, stream>>>(gkeys, bgk);
  pg_gemm<<<8, 256, 0, stream>>>(emb, ref, invn, sim);   // 64 waves * 64 cols = 4096
  dim3 rg(144, NLAB);                                    // 9*16 grid cells x 16 labels
  pg_reduce<<<rg, 256, 0, stream>>>(sim, thrp, gkeys, bgk);
  pg_final<<<NLAB, 256, 0, stream>>>(gkeys, bgk, out);
}